// LatentDenoisingLayer_1778116461097
// MI455X (gfx1250) — compile-verified
//
#include <hip/hip_runtime.h>
#include <hip/hip_bf16.h>

#define NN   20000
#define EE   200000
#define KK   9
#define CC   32
#define HH   8
#define DVV  8
#define DEE  64
#define DINN 128   // 2*CC + DEE

typedef __attribute__((ext_vector_type(2))) float v2f;
typedef __attribute__((ext_vector_type(8))) float v8f;

// ---- monotone float<->uint key for atomicMax-based segment max ----
__device__ __forceinline__ unsigned f2o(float f){
  unsigned b = __float_as_uint(f);
  return (b & 0x80000000u) ? ~b : (b | 0x80000000u);
}
__device__ __forceinline__ float o2f(unsigned u){
  return (u & 0x80000000u) ? __uint_as_float(u & 0x7fffffffu) : __uint_as_float(~u);
}

// workspace layout (in floats)
#define WS_XN     0          // N*K*C          = 5,760,000
#define WS_Z      5760000    // E*H            = 1,600,000
#define WS_SEGMAX 7360000    // N*H (as uint)  =   160,000
#define WS_DENOM  7520000    // N*H            =   160,000
#define WS_MSG    7680000    // N*K*64         = 11,520,000

__global__ void k_init(float* ws){
  int i = blockIdx.x*blockDim.x + threadIdx.x;
  float* msg = ws + WS_MSG;
  unsigned* segmax = (unsigned*)(ws + WS_SEGMAX);
  float* denom = ws + WS_DENOM;
  if (i < NN*KK*HH*DVV) msg[i] = 0.f;
  if (i < NN*HH){ segmax[i] = f2o(-3.0e38f); denom[i] = 0.f; }
}

__global__ void k_eqnorm1(const float* __restrict__ nf, const float* __restrict__ g,
                          float* __restrict__ xn){
  int i = blockIdx.x*blockDim.x + threadIdx.x;   // over N*C
  if (i >= NN*CC) return;
  int n = i >> 5, c = i & 31;
  float v[KK]; float s = 0.f;
  const float* p = nf + n*KK*CC + c;
  #pragma unroll
  for (int k=0;k<KK;k++){ v[k] = p[k*CC]; s += v[k]*v[k]; }
  float inv = rsqrtf(s*(1.0f/KK) + 1e-6f) * g[c];
  float* q = xn + n*KK*CC + c;
  #pragma unroll
  for (int k=0;k<KK;k++) q[k*CC] = v[k]*inv;
}

__global__ void k_logits(const float* __restrict__ xn, const float* __restrict__ ef,
                         const int* __restrict__ src, const int* __restrict__ dst,
                         const float* __restrict__ Wa, const float* __restrict__ ba,
                         float* __restrict__ zbuf, unsigned* __restrict__ segmax){
  __shared__ float Was[DINN*HH];
  __shared__ float bas[HH];
  for (int j = threadIdx.x; j < DINN*HH; j += blockDim.x) Was[j] = Wa[j];
  if (threadIdx.x < HH) bas[threadIdx.x] = ba[threadIdx.x];
  __syncthreads();
  int e = blockIdx.x*blockDim.x + threadIdx.x;
  if (e >= EE) return;
  int s = src[e], d = dst[e];
  const float* ps = xn + s*(KK*CC);     // xn[s,0,:]
  const float* pd = xn + d*(KK*CC);     // xn[d,0,:]
  const float* pe = ef + e*DEE;
  float acc[HH];
  #pragma unroll
  for (int h=0;h<HH;h++) acc[h] = bas[h];
  for (int j=0;j<CC;j++){
    float a = ps[j];
    #pragma unroll
    for (int h=0;h<HH;h++) acc[h] += a*Was[j*HH+h];
  }
  for (int j=0;j<CC;j++){
    float a = pd[j];
    #pragma unroll
    for (int h=0;h<HH;h++) acc[h] += a*Was[(CC+j)*HH+h];
  }
  for (int j=0;j<DEE;j++){
    float a = pe[j];
    #pragma unroll
    for (int h=0;h<HH;h++) acc[h] += a*Was[(2*CC+j)*HH+h];
  }
  #pragma unroll
  for (int h=0;h<HH;h++){
    zbuf[e*HH+h] = acc[h];
    atomicMax(&segmax[d*HH+h], f2o(acc[h]));
  }
}

__global__ void k_z(const int* __restrict__ dst, float* __restrict__ zbuf,
                    const unsigned* __restrict__ segmax, float* __restrict__ denom){
  int i = blockIdx.x*blockDim.x + threadIdx.x;
  if (i >= EE*HH) return;
  int e = i >> 3, h = i & 7;
  int d = dst[e];
  float m = o2f(segmax[d*HH+h]);
  float z = __expf(zbuf[i] - m);
  zbuf[i] = z;
  atomicAdd(&denom[d*HH+h], z);
}

// invert denominators once: rden = 1/(denom+1e-9)  (in place)
__global__ void k_rden(float* __restrict__ denom){
  int i = blockIdx.x*blockDim.x + threadIdx.x;
  if (i >= NN*HH) return;
  denom[i] = 1.0f / (denom[i] + 1e-9f);
}

// v = x_src @ Wv, scale by alpha, scatter-add into msg.  WMMA fp32, rows = (e,k) pairs.
__global__ void k_vmsg(const float* __restrict__ xn, const int* __restrict__ src,
                       const int* __restrict__ dst, const float* __restrict__ Wv,
                       const float* __restrict__ zbuf, const float* __restrict__ rden,
                       float* __restrict__ msg){
  __shared__ float As[16][CC];
  __shared__ int   rowE[16], rowK[16], rowD[16];
  __shared__ float alphaS[16][HH];
  int tid = threadIdx.x;
  int base = blockIdx.x * 16;               // flattened (e,k) row base; E*K % 16 == 0
  if (tid < 16){
    unsigned g = (unsigned)(base + tid);
    unsigned e = g / 9u;
    rowE[tid] = (int)e;
    rowK[tid] = (int)(g - e*9u);
    rowD[tid] = dst[e];
  }
  __syncthreads();
  for (int j = tid; j < 16*CC; j += blockDim.x){
    int r = j >> 5, c = j & 31;
    As[r][c] = xn[(src[rowE[r]]*KK + rowK[r])*CC + c];
  }
  {
    int r = tid >> 3, h = tid & 7;          // 128 threads -> 16x8 alpha tile
    alphaS[r][h] = zbuf[rowE[r]*HH + h] * rden[rowD[r]*HH + h];
  }
  __syncthreads();
  int wave = tid >> 5, lane = tid & 31;
  int m  = lane & 15;
  int kk = (lane >> 4) << 1;
  int n  = lane & 15;
  int ncol0 = wave * 16;
  v8f acc = {};
  #pragma unroll
  for (int k0 = 0; k0 < CC; k0 += 4){
    v2f a, b;
    a.x = As[m][k0 + kk];
    a.y = As[m][k0 + kk + 1];
    b.x = Wv[(k0 + kk)*64 + ncol0 + n];
    b.y = Wv[(k0 + kk + 1)*64 + ncol0 + n];
    acc = __builtin_amdgcn_wmma_f32_16x16x4_f32(false, a, false, b, (short)0, acc, false, false);
  }
  int rbase = (lane >> 4) * 8;
  int col = ncol0 + n;
  int h = col >> 3;
  #pragma unroll
  for (int i = 0; i < 8; i++){
    int r = rbase + i;
    atomicAdd(&msg[(rowD[r]*KK + rowK[r])*64 + col], acc[i] * alphaS[r][h]);
  }
}

// per-node block of 16 nodes = 144 (n,k) rows = 9 M-tiles; 9 waves, one tile each.
// attn = msg@Wo (WMMA) + residual; eqnorm2; gate; ffn1 = xn2@W1 * gate (WMMA);
// ffn2 = t@W2 (WMMA) + residual.
__global__ void __launch_bounds__(288)
k_node(const float* __restrict__ nf, const float* __restrict__ msg,
       const float* __restrict__ Wo, const float* __restrict__ ln2g,
       const float* __restrict__ Wg, const float* __restrict__ W1,
       const float* __restrict__ W2, float* __restrict__ node_out){
  __shared__ float B1[144*64];   // msg tile; later xn2 (first half) + t (second half)
  __shared__ float B2[144*32];   // x = node_feats + attn
  __shared__ float gateS[16*32];
  __shared__ int   rowNode[144];
  int tid = threadIdx.x;
  int nbase9 = blockIdx.x * 144;            // N % 16 == 0 -> exact
  // stage msg tile + row->node table
  if (tid < 144) rowNode[tid] = tid / 9;
  for (int j = tid; j < 144*64; j += 288) B1[j] = msg[nbase9*64 + j];
  __syncthreads();

  int w    = tid >> 5;                      // 0..8 -> M-tile
  int lane = tid & 31;
  int m  = lane & 15;
  int kk = (lane >> 4) << 1;
  int n  = lane & 15;
  int rbase = (lane >> 4) * 8;

  // ---- attn: (144x64) @ (64x32), 2 N-tiles ----
  #pragma unroll
  for (int nt = 0; nt < 2; nt++){
    int ncol0 = nt * 16;
    v8f acc = {};
    #pragma unroll
    for (int k0 = 0; k0 < 64; k0 += 4){
      v2f a, b;
      a.x = B1[(w*16 + m)*64 + k0 + kk];
      a.y = B1[(w*16 + m)*64 + k0 + kk + 1];
      b.x = Wo[(k0 + kk)*CC + ncol0 + n];
      b.y = Wo[(k0 + kk + 1)*CC + ncol0 + n];
      acc = __builtin_amdgcn_wmma_f32_16x16x4_f32(false, a, false, b, (short)0, acc, false, false);
    }
    #pragma unroll
    for (int i = 0; i < 8; i++){
      int row = w*16 + rbase + i;
      int col = ncol0 + n;
      B2[row*CC + col] = acc[i] + nf[(nbase9 + row)*CC + col];
    }
  }
  __syncthreads();

  // ---- eqnorm2 per (node, c): write xn2 into B1[0 .. 144*32) ----
  for (int idx = tid; idx < 16*CC; idx += 288){
    int j = idx >> 5, c = idx & 31;
    float s = 0.f;
    #pragma unroll
    for (int k = 0; k < KK; k++){
      float v = B2[(j*KK + k)*CC + c];
      s += v*v;
    }
    float inv = rsqrtf(s*(1.0f/KK) + 1e-6f) * ln2g[c];
    #pragma unroll
    for (int k = 0; k < KK; k++)
      B1[(j*KK + k)*CC + c] = B2[(j*KK + k)*CC + c] * inv;
  }
  __syncthreads();

  // ---- gate = silu(xn2[:,0,:] @ Wg) per (node, c) ----
  for (int idx = tid; idx < 16*CC; idx += 288){
    int j = idx >> 5, c = idx & 31;
    float g = 0.f;
    for (int c2 = 0; c2 < CC; c2++) g += B1[(j*KK)*CC + c2] * Wg[c2*CC + c];
    gateS[idx] = g / (1.0f + __expf(-g));
  }
  __syncthreads();

  // ---- ffn1: (144x32) @ (32x32), times gate; t -> B1[144*32 .. 144*64) ----
  float* xn2S = B1;
  float* tS   = B1 + 144*CC;
  #pragma unroll
  for (int nt = 0; nt < 2; nt++){
    int ncol0 = nt * 16;
    v8f acc = {};
    #pragma unroll
    for (int k0 = 0; k0 < CC; k0 += 4){
      v2f a, b;
      a.x = xn2S[(w*16 + m)*CC + k0 + kk];
      a.y = xn2S[(w*16 + m)*CC + k0 + kk + 1];
      b.x = W1[(k0 + kk)*CC + ncol0 + n];
      b.y = W1[(k0 + kk + 1)*CC + ncol0 + n];
      acc = __builtin_amdgcn_wmma_f32_16x16x4_f32(false, a, false, b, (short)0, acc, false, false);
    }
    #pragma unroll
    for (int i = 0; i < 8; i++){
      int row = w*16 + rbase + i;
      int col = ncol0 + n;
      tS[row*CC + col] = acc[i] * gateS[rowNode[row]*CC + col];
    }
  }
  __syncthreads();

  // ---- ffn2: (144x32) @ (32x32); node_out = x + ffn2 ----
  #pragma unroll
  for (int nt = 0; nt < 2; nt++){
    int ncol0 = nt * 16;
    v8f acc = {};
    #pragma unroll
    for (int k0 = 0; k0 < CC; k0 += 4){
      v2f a, b;
      a.x = tS[(w*16 + m)*CC + k0 + kk];
      a.y = tS[(w*16 + m)*CC + k0 + kk + 1];
      b.x = W2[(k0 + kk)*CC + ncol0 + n];
      b.y = W2[(k0 + kk + 1)*CC + ncol0 + n];
      acc = __builtin_amdgcn_wmma_f32_16x16x4_f32(false, a, false, b, (short)0, acc, false, false);
    }
    #pragma unroll
    for (int i = 0; i < 8; i++){
      int row = w*16 + rbase + i;
      int col = ncol0 + n;
      node_out[(nbase9 + row)*CC + col] = B2[row*CC + col] + acc[i];
    }
  }
}

// edge MLP: relu(inp@Wf1+bf1)@Wf2+bf2, LayerNorm, residual.  WMMA fp32, 16 edges/block.
__global__ void k_edge(const float* __restrict__ node_out, const float* __restrict__ ef,
                       const int* __restrict__ src, const int* __restrict__ dst,
                       const float* __restrict__ Wf1, const float* __restrict__ bf1,
                       const float* __restrict__ Wf2, const float* __restrict__ bf2,
                       const float* __restrict__ lng, const float* __restrict__ lnb,
                       float* __restrict__ edge_out){
  __shared__ float As[16][DINN];
  __shared__ float H1[16][DEE];
  __shared__ float U[16][DEE];
  __shared__ float MU[16], RS[16];
  int tid = threadIdx.x;
  int ebase = blockIdx.x * 16;              // E % 16 == 0
  for (int j = tid; j < 16*DINN; j += blockDim.x){
    int r = j >> 7, col = j & 127;
    int e = ebase + r;
    float v;
    if (col < 32)      v = node_out[src[e]*KK*CC + col];
    else if (col < 64) v = node_out[dst[e]*KK*CC + (col-32)];
    else               v = ef[e*DEE + (col-64)];
    As[r][col] = v;
  }
  __syncthreads();
  int wave = tid >> 5, lane = tid & 31;
  int m  = lane & 15;
  int kk = (lane >> 4) << 1;
  int n  = lane & 15;
  int ncol0 = wave * 16;
  v8f acc = {};
  #pragma unroll
  for (int k0 = 0; k0 < DINN; k0 += 4){
    v2f a, b;
    a.x = As[m][k0+kk]; a.y = As[m][k0+kk+1];
    b.x = Wf1[(k0+kk)*DEE + ncol0 + n];
    b.y = Wf1[(k0+kk+1)*DEE + ncol0 + n];
    acc = __builtin_amdgcn_wmma_f32_16x16x4_f32(false, a, false, b, (short)0, acc, false, false);
  }
  int rbase = (lane >> 4) * 8;
  int col = ncol0 + n;
  #pragma unroll
  for (int i=0;i<8;i++)
    H1[rbase+i][col] = fmaxf(acc[i] + bf1[col], 0.f);
  __syncthreads();
  v8f acc2 = {};
  #pragma unroll
  for (int k0 = 0; k0 < DEE; k0 += 4){
    v2f a, b;
    a.x = H1[m][k0+kk]; a.y = H1[m][k0+kk+1];
    b.x = Wf2[(k0+kk)*DEE + ncol0 + n];
    b.y = Wf2[(k0+kk+1)*DEE + ncol0 + n];
    acc2 = __builtin_amdgcn_wmma_f32_16x16x4_f32(false, a, false, b, (short)0, acc2, false, false);
  }
  #pragma unroll
  for (int i=0;i<8;i++)
    U[rbase+i][col] = acc2[i] + bf2[col];
  __syncthreads();
  if (tid < 16){
    float mu = 0.f;
    for (int j=0;j<DEE;j++) mu += U[tid][j];
    mu *= (1.0f/DEE);
    float var = 0.f;
    for (int j=0;j<DEE;j++){ float d = U[tid][j]-mu; var += d*d; }
    var *= (1.0f/DEE);
    MU[tid] = mu; RS[tid] = rsqrtf(var + 1e-5f);
  }
  __syncthreads();
  for (int j = tid; j < 16*DEE; j += blockDim.x){
    int r = j >> 6, c2 = j & 63;
    int e = ebase + r;
    float val = (U[r][c2]-MU[r])*RS[r]*lng[c2] + lnb[c2];
    edge_out[e*DEE + c2] = ef[e*DEE + c2] + val;
  }
}

extern "C" void kernel_launch(void* const* d_in, const int* in_sizes, int n_in,
                              void* d_out, int out_size, void* d_ws, size_t ws_size,
                              hipStream_t stream){
  const float* nf  = (const float*)d_in[0];
  const float* ef  = (const float*)d_in[1];
  const int*   ei  = (const int*)  d_in[2];
  const float* Wa  = (const float*)d_in[3];
  const float* ba  = (const float*)d_in[4];
  const float* Wv  = (const float*)d_in[5];
  const float* Wo  = (const float*)d_in[6];
  const float* ln1g= (const float*)d_in[7];
  const float* ln2g= (const float*)d_in[8];
  const float* Wg  = (const float*)d_in[9];
  const float* W1  = (const float*)d_in[10];
  const float* W2  = (const float*)d_in[11];
  const float* Wf1 = (const float*)d_in[12];
  const float* bf1 = (const float*)d_in[13];
  const float* Wf2 = (const float*)d_in[14];
  const float* bf2 = (const float*)d_in[15];
  const float* lng = (const float*)d_in[16];
  const float* lnb = (const float*)d_in[17];
  const int* src = ei;
  const int* dst = ei + EE;
  float* ws   = (float*)d_ws;
  float* xn       = ws + WS_XN;
  float* zbuf     = ws + WS_Z;
  unsigned* segmx = (unsigned*)(ws + WS_SEGMAX);
  float* denom    = ws + WS_DENOM;
  float* msg      = ws + WS_MSG;
  float* node_out = (float*)d_out;
  float* edge_out = node_out + NN*KK*CC;

  k_init   <<<(NN*KK*64 + 255)/256, 256, 0, stream>>>(ws);
  k_eqnorm1<<<(NN*CC + 255)/256,    256, 0, stream>>>(nf, ln1g, xn);
  k_logits <<<(EE + 255)/256,       256, 0, stream>>>(xn, ef, src, dst, Wa, ba, zbuf, segmx);
  k_z      <<<(EE*HH + 255)/256,    256, 0, stream>>>(dst, zbuf, segmx, denom);
  k_rden   <<<(NN*HH + 255)/256,    256, 0, stream>>>(denom);
  k_vmsg   <<<EE*KK/16,             128, 0, stream>>>(xn, src, dst, Wv, zbuf, denom, msg);
  k_node   <<<NN/16,                288, 0, stream>>>(nf, msg, Wo, ln2g, Wg, W1, W2, node_out);
  k_edge   <<<EE/16,                128, 0, stream>>>(node_out, ef, src, dst, Wf1, bf1,
                                                      Wf2, bf2, lng, lnb, edge_out);
}